// LTC_15874199126391
// MI455X (gfx1250) — compile-verified
//
#include <hip/hip_runtime.h>
#include <stdint.h>

// ---------------------------------------------------------------------------
// LTC forward for MI455X (gfx1250).
//   B=256, L=1024, C=64, N=128, ODE_UNFOLDS=6.
// Per-edge nonlinearity => no GEMM structure => no WMMA. Instead we use the
// CDNA5 paths the math rewards: tensor_load_to_lds (TDM) weight staging,
// global_load_async_to_lds_b128 x prefetch, v_tanh_f32, v_fma_mix_f32_bf16.
// ---------------------------------------------------------------------------

#define N_UNITS 128
#define IN_DIM  64
#define SEQ_L   1024
#define BATCH   256
#define UNFOLDS 6

// LDS layout (dynamic shared only => base offset 0, used by TDM descriptors)
#define REC_OFF   0                         // 128*128 * uint2 = 131072 B
#define SEN_OFF   131072                    //  64*128 * uint2 =  65536 B
#define V_OFF     (131072 + 65536)          // 2buf * 2batch * 128 f32 = 2048 B (4096 B reserved)
#define INP_OFF   (V_OFF + 4096)            // 2buf * 2batch *  64 f32 = 1024 B (2048 B reserved)
#define LDS_BYTES (INP_OFF + 2048)          // 202752 B  (< 320 KB / WGP)

__device__ __forceinline__ unsigned bf16u(float f) {
  unsigned u = __float_as_uint(f);
  return (u + 0x7fffu + ((u >> 16) & 1u)) >> 16;   // round-to-nearest-even
}
__device__ __forceinline__ float softplusf(float x) { return log1pf(__expf(x)); }

__device__ __forceinline__ float tanh_fast(float x) {
#if __has_builtin(__builtin_amdgcn_tanhf)
  return __builtin_amdgcn_tanhf(x);
#else
  float r;
  asm("v_tanh_f32 %0, %1" : "=v"(r) : "v"(x));
  return r;
#endif
}

// One synapse edge.  wgt.x = {hi: bf16(-0.5*mu*sigma), lo: bf16(0.5*sigma)}
//                    wgt.y = {hi: bf16(0.5*wp),        lo: bf16(0.5*wp*erev)}
// den += 0.5*wp      * tanh(0.5*sigma*v - 0.5*mu*sigma)
// num += 0.5*wp*erev * tanh(...)
// (the constant +0.5*wp terms are hoisted into per-neuron base sums)
__device__ __forceinline__ void ltc_edge(float vi, uint2 wgt, float& num, float& den) {
  float arg, th;
  asm("v_fma_mix_f32_bf16 %0, %1, %2, %2 op_sel:[0,0,1] op_sel_hi:[0,1,1]"
      : "=v"(arg) : "v"(vi), "v"(wgt.x));
  th = tanh_fast(arg);
  asm("v_fma_mix_f32_bf16 %0, %1, %2, %0 op_sel:[1,0,0] op_sel_hi:[1,0,0]"
      : "+v"(den) : "v"(wgt.y), "v"(th));
  asm("v_fma_mix_f32_bf16 %0, %1, %2, %0 op_sel:[0,0,0] op_sel_hi:[1,0,0]"
      : "+v"(num) : "v"(wgt.y), "v"(th));
}

// ---------------------------------------------------------------------------
// Prep kernel: fold softplus / halving / input-affine into bf16-packed weights
// ---------------------------------------------------------------------------
__global__ void ltc_pack(const float* __restrict__ sigma, const float* __restrict__ mu,
                         const float* __restrict__ w,     const float* __restrict__ erev,
                         const float* __restrict__ ssig,  const float* __restrict__ smu,
                         const float* __restrict__ sw,    const float* __restrict__ serev,
                         const float* __restrict__ iw,    const float* __restrict__ ib,
                         uint2* __restrict__ recPK, uint2* __restrict__ senPK) {
  int idx = blockIdx.x * blockDim.x + threadIdx.x;
  if (idx < N_UNITS * N_UNITS) {
    float sg = sigma[idx], m = mu[idx], wp = softplusf(w[idx]), er = erev[idx];
    float A = 0.5f * sg, nBc = -0.5f * m * sg, wph = 0.5f * wp, wer = wph * er;
    uint2 d;
    d.x = (bf16u(nBc) << 16) | bf16u(A);
    d.y = (bf16u(wph) << 16) | bf16u(wer);
    recPK[idx] = d;
  } else if (idx < N_UNITS * N_UNITS + IN_DIM * N_UNITS) {
    int s = idx - N_UNITS * N_UNITS;
    int c = s >> 7;
    float sg = ssig[s], m = smu[s], wp = softplusf(sw[s]), er = serev[s];
    float iwc = iw[c], ibc = ib[c];
    // (x*iw + ib - mu)*sigma  ->  x*(iw*sigma) + (ib-mu)*sigma   (halved)
    float A = 0.5f * iwc * sg, nBc = 0.5f * (ibc - m) * sg, wph = 0.5f * wp, wer = wph * er;
    uint2 d;
    d.x = (bf16u(nBc) << 16) | bf16u(A);
    d.y = (bf16u(wph) << 16) | bf16u(wer);
    senPK[s] = d;
  }
}

// ---------------------------------------------------------------------------
// TDM descriptor (D#) for a simple 2D row-major tile copy into LDS
// ---------------------------------------------------------------------------
typedef unsigned int u32x4 __attribute__((ext_vector_type(4)));
typedef int          i32x8 __attribute__((ext_vector_type(8)));
typedef int          i32x4 __attribute__((ext_vector_type(4)));

__device__ __forceinline__ void tdm_load_2d(unsigned lds_off, const void* gptr,
                                            unsigned rowElems, unsigned rows,
                                            unsigned dsize_code /*3 = 8B elems*/) {
  unsigned long long ga = (unsigned long long)(uintptr_t)gptr;
  u32x4 g0;
  i32x8 g1;
  i32x4 g2 = {0, 0, 0, 0}, g3 = {0, 0, 0, 0};
  i32x8 g4 = {0, 0, 0, 0, 0, 0, 0, 0};
  g0[0] = 1u;                                               // count=1, user mode
  g0[1] = lds_off;                                          // lds_addr
  g0[2] = (unsigned)(ga & 0xffffffffu);                     // global_addr[31:0]
  g0[3] = (unsigned)((ga >> 32) & 0x01ffffffu) | (2u << 30);// addr[56:32] | type=2
  g1[0] = (int)(dsize_code << 16);                          // wg_mask=0 | data_size
  g1[1] = (int)((rowElems & 0xffffu) << 16);                // tensor_dim0 lo16
  g1[2] = (int)(((rowElems >> 16) & 0xffffu) | ((rows & 0xffffu) << 16)); // dim0 hi | dim1 lo
  g1[3] = (int)(((rows >> 16) & 0xffffu) | ((rowElems & 0xffffu) << 16)); // dim1 hi | tile_dim0
  g1[4] = (int)(rows & 0xffffu);                            // tile_dim1 | tile_dim2=0
  g1[5] = (int)rowElems;                                    // tensor_dim0_stride lo32
  g1[6] = 0;
  g1[7] = 0;
  // 6-arg toolchain variant: (g0, g1, g2, g3, g4, cpol)
  __builtin_amdgcn_tensor_load_to_lds(g0, g1, g2, g3, g4, 0);
}

// ---------------------------------------------------------------------------
// Main persistent kernel: 128 WGs x 256 threads; WG owns 2 batch rows,
// thread = (batch half, neuron n). Entire L=1024 sequence in one launch.
// ---------------------------------------------------------------------------
__global__ void __launch_bounds__(256)
ltc_main(const float* __restrict__ x,
         const float* __restrict__ gleak, const float* __restrict__ vleakp,
         const float* __restrict__ cmv,
         const float* __restrict__ outw,  const float* __restrict__ outb,
         const uint2* __restrict__ recPKg, const uint2* __restrict__ senPKg,
         float* __restrict__ readout, float* __restrict__ vfinal) {
  extern __shared__ char smem[];
  uint2* recL = (uint2*)(smem + REC_OFF);
  uint2* senL = (uint2*)(smem + SEN_OFF);
  float* vL   = (float*)(smem + V_OFF);    // [buf][batch][128]
  float* xL   = (float*)(smem + INP_OFF);  // [buf][batch][64]

  const int tid = threadIdx.x;
  const int b   = tid >> 7;        // batch half within WG
  const int n   = tid & 127;       // neuron
  const int bg  = blockIdx.x * 2 + b;

  // --- stage packed weights into LDS via the Tensor Data Mover ---
  if (tid == 0) {
    tdm_load_2d(REC_OFF, recPKg, N_UNITS, N_UNITS, 3);
    tdm_load_2d(SEN_OFF, senPKg, N_UNITS, IN_DIM, 3);
  }
  __builtin_amdgcn_s_wait_tensorcnt(0);
  __syncthreads();

  // --- per-neuron scalar params (registers) ---
  float gl   = softplusf(gleak[n]);
  float glvl = gl * vleakp[n];
  float cmt  = (float)UNFOLDS * softplusf(cmv[n]);   // softplus(cm) / (1/6)
  float ow = outw[n], ob = outb[n];

  // base sums of the hoisted +0.5*wp (and *erev) terms, computed from the
  // SAME bf16 values used per-edge (consistency with the packed weights)
  float rnb = 0.f, rdb = 0.f;
  for (int i = 0; i < N_UNITS; ++i) {
    unsigned d1 = recL[(i << 7) | n].y;
    rdb += __uint_as_float(d1 & 0xffff0000u);
    rnb += __uint_as_float(d1 << 16);
  }
  float snb = 0.f, sdb = 0.f;
  for (int c = 0; c < IN_DIM; ++c) {
    unsigned d1 = senL[(c << 7) | n].y;
    sdb += __uint_as_float(d1 & 0xffff0000u);
    snb += __uint_as_float(d1 << 16);
  }

  // v init (buffer 0)
  vL[b * 128 + n] = 0.f;
  float myv = 0.f;

  // --- async prefetch x[t=0] into x-buffer 0 (waves 0 & 4, 16 lanes each) ---
  {
    int l = tid & 127;
    if (l < 16) {
      unsigned lo = (unsigned)(INP_OFF + b * 256 + l * 16);
      const float* gp = x + (size_t)bg * SEQ_L * IN_DIM + l * 4;
      asm volatile("global_load_async_to_lds_b128 %0, %1, off"
                   :: "v"(lo), "v"(gp) : "memory");
    }
  }

  int cur = 0;  // v double-buffer index (flips 6x per step -> invariant per t)
  for (int t = 0; t < SEQ_L; ++t) {
    asm volatile("s_wait_asynccnt 0x0" ::: "memory");
    __syncthreads();
    const int xb = t & 1;

    // prefetch next timestep's x while we compute this one
    if (t + 1 < SEQ_L) {
      int l = tid & 127;
      if (l < 16) {
        unsigned lo = (unsigned)(INP_OFF + ((t + 1) & 1) * 512 + b * 256 + l * 16);
        const float* gp = x + ((size_t)bg * SEQ_L + (t + 1)) * IN_DIM + l * 4;
        asm volatile("global_load_async_to_lds_b128 %0, %1, off"
                     :: "v"(lo), "v"(gp) : "memory");
      }
    }

    // --- sensory synapses (input affine folded into weights) ---
    float ns0 = snb, nd0 = sdb, ns1 = 0.f, nd1 = 0.f;
    const float* xrow = xL + xb * 128 + b * 64;
#pragma unroll 4
    for (int c = 0; c < IN_DIM; c += 2) {
      ltc_edge(xrow[c],     senL[(c << 7) | n],       ns0, nd0);
      ltc_edge(xrow[c + 1], senL[((c + 1) << 7) | n], ns1, nd1);
    }
    const float num_s = ns0 + ns1, den_s = nd0 + nd1;

    // --- fused ODE unfolds ---
#pragma unroll 1
    for (int u = 0; u < UNFOLDS; ++u) {
      float n0 = rnb + num_s, d0 = rdb + den_s, n1 = 0.f, d1 = 0.f;
      const float* vrow = vL + cur * 256 + b * 128;
#pragma unroll 4
      for (int i = 0; i < N_UNITS; i += 2) {
        ltc_edge(vrow[i],     recL[(i << 7) | n],       n0, d0);
        ltc_edge(vrow[i + 1], recL[((i + 1) << 7) | n], n1, d1);
      }
      float num = n0 + n1, den = d0 + d1;
      myv = (cmt * myv + glvl + num) / (cmt + gl + den + 1e-8f);
      vL[(cur ^ 1) * 256 + b * 128 + n] = myv;
      __syncthreads();
      cur ^= 1;
    }

    readout[((size_t)bg * SEQ_L + t) * N_UNITS + n] = fmaf(myv, ow, ob);
  }
  vfinal[bg * N_UNITS + n] = myv;
}

// ---------------------------------------------------------------------------
extern "C" void kernel_launch(void* const* d_in, const int* in_sizes, int n_in,
                              void* d_out, int out_size, void* d_ws, size_t ws_size,
                              hipStream_t stream) {
  (void)in_sizes; (void)n_in; (void)out_size; (void)ws_size;

  const float* x     = (const float*)d_in[0];
  const float* gleak = (const float*)d_in[1];
  const float* vleak = (const float*)d_in[2];
  const float* cm    = (const float*)d_in[3];
  const float* sigma = (const float*)d_in[4];
  const float* mu    = (const float*)d_in[5];
  const float* w     = (const float*)d_in[6];
  const float* erev  = (const float*)d_in[7];
  const float* ssig  = (const float*)d_in[8];
  const float* smu   = (const float*)d_in[9];
  const float* sw    = (const float*)d_in[10];
  const float* serev = (const float*)d_in[11];
  const float* iw    = (const float*)d_in[12];
  const float* ib    = (const float*)d_in[13];
  const float* ow    = (const float*)d_in[14];
  const float* ob    = (const float*)d_in[15];

  uint2* recPK = (uint2*)d_ws;                                          // 131072 B
  uint2* senPK = (uint2*)((char*)d_ws + (size_t)N_UNITS * N_UNITS * 8); //  65536 B

  const int packTotal = N_UNITS * N_UNITS + IN_DIM * N_UNITS;  // 24576
  ltc_pack<<<packTotal / 256, 256, 0, stream>>>(sigma, mu, w, erev,
                                                ssig, smu, sw, serev,
                                                iw, ib, recPK, senPK);

  float* readout = (float*)d_out;
  float* vfin    = readout + (size_t)BATCH * SEQ_L * N_UNITS;
  ltc_main<<<BATCH / 2, 256, LDS_BYTES, stream>>>(x, gleak, vleak, cm, ow, ob,
                                                  recPK, senPK, readout, vfin);
}